// ObjectCondensationLoss_63926293234225
// MI455X (gfx1250) — compile-verified
//
#include <hip/hip_runtime.h>
#include <math.h>

#define BB   8
#define NN   200000
#define DD   16
#define NI   256
#define NSEG (NI + 1)

typedef __attribute__((ext_vector_type(2))) float v2f;
typedef __attribute__((ext_vector_type(8))) float v8f;

__device__ __forceinline__ float softplusf(float x) {
    // stable log1p(exp(x))
    return fmaxf(x, 0.0f) + log1pf(expf(-fabsf(x)));
}

// ---------------------------------------------------------------------------
// Init: first_cp <- N, everything else <- 0
// ---------------------------------------------------------------------------
__global__ void oc_init(int* __restrict__ firstcp, int* __restrict__ posCnt,
                        int* __restrict__ compactCnt, float* __restrict__ sums,
                        float* __restrict__ segsum, float* __restrict__ segcnt,
                        float* __restrict__ repsum, float* __restrict__ norms,
                        float* __restrict__ cpEmb) {
    int idx = blockIdx.x * blockDim.x + threadIdx.x;
    if (idx < BB * NSEG) { firstcp[idx] = NN; segsum[idx] = 0.0f; segcnt[idx] = 0.0f; }
    if (idx < BB * 4)    { sums[idx] = 0.0f; }
    if (idx < BB)        { posCnt[idx] = 0; compactCnt[idx] = 0; repsum[idx] = 0.0f; }
    if (idx < BB * NI)   { norms[idx] = 0.0f; }
    if (idx < BB * NI * DD) { cpEmb[idx] = 0.0f; }
}

// ---------------------------------------------------------------------------
// Pass 1: BCE partial sums (Scp, Snoncp, Sbg), pos count, first_cp (atomicMin),
//         cp compaction into dense cpEmb + per-row squared norms.
// ---------------------------------------------------------------------------
__global__ void oc_pass1(const float* __restrict__ beta,
                         const float* __restrict__ embed,
                         const int*   __restrict__ sid,
                         const int*   __restrict__ iscp,
                         int* __restrict__ firstcp, int* __restrict__ posCnt,
                         int* __restrict__ compactCnt, float* __restrict__ sums,
                         float* __restrict__ cpEmb, float* __restrict__ norms) {
    const int b = blockIdx.y;
    const int i = blockIdx.x * blockDim.x + threadIdx.x;

    float scp = 0.0f, sn = 0.0f, sb = 0.0f;
    int p = 0;
    if (i < NN) {
        const size_t gi = (size_t)b * NN + i;
        const float x  = beta[gi];
        const int   s  = sid[gi];
        const bool  cp = (iscp[gi] == 1);
        if (cp) {
            scp = softplusf(-x);
            p = 1;
            const int sc = (s >= 0) ? s : NI;
            atomicMin(&firstcp[b * NSEG + sc], i);
            const int k = atomicAdd(&compactCnt[b], 1);
            if (k < NI) {
                const float* e = embed + gi * DD;
                float* dst = cpEmb + ((size_t)b * NI + k) * DD;
                float nrm = 0.0f;
#pragma unroll
                for (int d = 0; d < DD; ++d) { float v = e[d]; dst[d] = v; nrm += v * v; }
                norms[b * NI + k] = nrm;
            }
        } else if (s >= 0) {
            sn = softplusf(x);
        } else {
            sb = softplusf(x);
        }
    }

    __shared__ float r0[256], r1[256], r2[256];
    __shared__ int   ri[256];
    const int t = threadIdx.x;
    r0[t] = scp; r1[t] = sn; r2[t] = sb; ri[t] = p;
    __syncthreads();
    for (int o = 128; o > 0; o >>= 1) {
        if (t < o) { r0[t] += r0[t + o]; r1[t] += r1[t + o]; r2[t] += r2[t + o]; ri[t] += ri[t + o]; }
        __syncthreads();
    }
    if (t == 0) {
        atomicAdd(&sums[b * 4 + 0], r0[0]);
        atomicAdd(&sums[b * 4 + 1], r1[0]);
        atomicAdd(&sums[b * 4 + 2], r2[0]);
        atomicAdd(&posCnt[b], ri[0]);
    }
}

// ---------------------------------------------------------------------------
// Attraction: per-point squared distance to its segment's first-cp embedding.
// first_cp table staged in LDS; LDS-staged segment accumulation, flushed once
// per block.
// ---------------------------------------------------------------------------
__global__ void oc_attr(const float* __restrict__ embed,
                        const int*   __restrict__ sid,
                        const int*   __restrict__ firstcp,
                        float* __restrict__ segsum, float* __restrict__ segcnt) {
    const int b = blockIdx.y;
    __shared__ float lsum[NSEG];
    __shared__ float lcnt[NSEG];
    __shared__ int   lfc[NSEG];
    for (int s = threadIdx.x; s < NSEG; s += blockDim.x) {
        lsum[s] = 0.0f;
        lcnt[s] = 0.0f;
        lfc[s]  = firstcp[b * NSEG + s];
    }
    __syncthreads();

    const int stride = gridDim.x * blockDim.x;
    for (int i = blockIdx.x * blockDim.x + threadIdx.x; i < NN; i += stride) {
        const int s = sid[(size_t)b * NN + i];
        if (s >= 0) {
            const int fc = lfc[s];
            if (fc < NN) {  // has_cp: only these segments contribute to attr
                const float* e = embed + ((size_t)b * NN + i) * DD;
                const float* c = embed + ((size_t)b * NN + fc) * DD;
                float d2 = 0.0f;
#pragma unroll
                for (int d = 0; d < DD; d += 4) {
                    float4 ev = *(const float4*)(e + d);
                    float4 cv = *(const float4*)(c + d);
                    float dx = ev.x - cv.x, dy = ev.y - cv.y;
                    float dz = ev.z - cv.z, dw = ev.w - cv.w;
                    d2 += dx * dx + dy * dy + dz * dz + dw * dw;
                }
                atomicAdd(&lsum[s], d2);
                atomicAdd(&lcnt[s], 1.0f);
            }
        }
    }
    __syncthreads();
    for (int s = threadIdx.x; s < NSEG; s += blockDim.x) {
        if (lcnt[s] != 0.0f) {
            atomicAdd(&segsum[b * NSEG + s], lsum[s]);
            atomicAdd(&segcnt[b * NSEG + s], lcnt[s]);
        }
    }
}

// ---------------------------------------------------------------------------
// Repulsion: 256x256 Gram matrix per batch via V_WMMA_F32_16X16X4_F32.
// d2(i,j) = ||e_i||^2 + ||e_j||^2 - 2 * <e_i, e_j>; accumulate exp(-d2).
// One wave per 16x16 output tile; K=16 via 4 chained WMMAs.
// A (16x4) and B (4x16) f32 operands: lane l, vgpr v -> [row=l%16][k=v+2*(l>=16)]
// Masking is done by forcing d2 = +inf (exp(-inf) == 0) so the expf stays
// branch-free (no exec-mask scalarization).
// ---------------------------------------------------------------------------
__global__ void oc_rep(const float* __restrict__ cpEmb,
                       const float* __restrict__ norms,
                       const int*   __restrict__ posCnt,
                       float* __restrict__ repsum) {
    const int b    = blockIdx.y;
    const int wave = threadIdx.x >> 5;
    const int lane = threadIdx.x & 31;
    const int tile = blockIdx.x * (blockDim.x >> 5) + wave;  // 0..255
    const int m0 = (tile >> 4) << 4;
    const int n0 = (tile & 15) << 4;

    int K = posCnt[b];
    if (K > NI) K = NI;

    const int hi = lane >> 4;   // 0: k in {0,1}; 1: k in {2,3}
    const int r  = lane & 15;
    const float* Arow = cpEmb + ((size_t)b * NI + m0 + r) * DD + 2 * hi;
    const float* Brow = cpEmb + ((size_t)b * NI + n0 + r) * DD + 2 * hi;

    v8f acc = {};
#pragma unroll
    for (int c = 0; c < 4; ++c) {
        v2f a  = *(const v2f*)(Arow + 4 * c);
        v2f bb = *(const v2f*)(Brow + 4 * c);
        // 8 args: (neg_a, A, neg_b, B, c_mod, C, reuse_a, reuse_b)
        acc = __builtin_amdgcn_wmma_f32_16x16x4_f32(false, a, false, bb,
                                                    (short)0, acc, false, false);
    }

    const int   n     = n0 + r;
    const float nn    = norms[b * NI + n];
    const bool  nok   = (n < K);
    const float inf   = __builtin_inff();

    // Unconditional norm loads for all 8 row indices of this lane's column.
    float nm[8];
#pragma unroll
    for (int vr = 0; vr < 8; ++vr)
        nm[vr] = norms[b * NI + m0 + vr + hi * 8];

    float partial = 0.0f;
#pragma unroll
    for (int vr = 0; vr < 8; ++vr) {
        const int  m    = m0 + vr + hi * 8;  // C/D layout: vgpr r -> M=r (+8 for hi)
        const bool mask = (m < K) && nok;
        float d2 = nm[vr] + nn - 2.0f * acc[vr];
        d2 = mask ? d2 : inf;                // exp(-inf) == 0, keeps expf branch-free
        partial += expf(-d2);
    }
#pragma unroll
    for (int off = 16; off > 0; off >>= 1)
        partial += __shfl_xor(partial, off, 32);
    if (lane == 0) atomicAdd(&repsum[b], partial);
}

// ---------------------------------------------------------------------------
// Finalize: per-batch beta_loss / attr / rep, validity mask, averages.
// ---------------------------------------------------------------------------
__global__ void oc_final(const float* __restrict__ sums,
                         const int*   __restrict__ posCnt,
                         const int*   __restrict__ firstcp,
                         const float* __restrict__ segsum,
                         const float* __restrict__ segcnt,
                         const float* __restrict__ repsum,
                         float* __restrict__ out) {
    __shared__ float red[256];
    __shared__ float bl[BB], at[BB], rp[BB];
    __shared__ int   vld[BB];
    const int t = threadIdx.x;

    for (int b = 0; b < BB; ++b) {
        float v = 0.0f;
        if (t < NI) {
            const int fc = firstcp[b * NSEG + t];
            if (fc < NN) {
                const float cnt = segcnt[b * NSEG + t];
                if (cnt > 0.0f) v = segsum[b * NSEG + t] / fmaxf(cnt, 1.0f);
            }
        }
        red[t] = v;
        __syncthreads();
        for (int o = 128; o > 0; o >>= 1) { if (t < o) red[t] += red[t + o]; __syncthreads(); }
        if (t == 0) at[b] = red[0];  // ATTR_W = 1.0
        __syncthreads();
    }

    if (t < BB) {
        const int   b   = t;
        const int   pos = posCnt[b];
        const float neg = (float)NN - (float)pos;
        const float pw  = neg / ((float)pos + 1e-6f);
        bl[b] = (pw * sums[b * 4 + 0] + sums[b * 4 + 1] + 2.0f * sums[b * 4 + 2]) / (float)NN;
        const int   Kc = (pos < NI) ? pos : NI;
        const float pm = fmaxf((float)Kc * (float)Kc, 1.0f);
        rp[b]  = (pos > 1) ? (repsum[b] / pm) : 0.0f;  // REPL_W = 1.0
        vld[b] = (pos >= 1 && neg >= 1.0f) ? 1 : 0;
    }
    __syncthreads();

    if (t == 0) {
        float cnt = 0.0f, f = 0.0f, sbl = 0.0f, sat = 0.0f, srp = 0.0f;
        for (int b = 0; b < BB; ++b) {
            const float vf = (float)vld[b];
            cnt += vf;
            sbl += bl[b] * vf;
            sat += at[b] * vf;
            srp += rp[b] * vf;
            f   += (bl[b] + at[b] + rp[b]) * vf;
        }
        cnt = fmaxf(cnt, 1.0f);
        out[0] = f / cnt;
        out[1] = sbl / cnt;
        out[2] = sat / cnt;
        out[3] = srp / cnt;
    }
}

// ---------------------------------------------------------------------------
extern "C" void kernel_launch(void* const* d_in, const int* in_sizes, int n_in,
                              void* d_out, int out_size, void* d_ws, size_t ws_size,
                              hipStream_t stream) {
    (void)in_sizes; (void)n_in; (void)out_size; (void)ws_size;
    const float* beta  = (const float*)d_in[0];
    const float* embed = (const float*)d_in[1];
    const int*   sid   = (const int*)d_in[2];
    const int*   iscp  = (const int*)d_in[3];
    float* out = (float*)d_out;

    char* ws = (char*)d_ws;
    size_t off = 0;
    auto carve = [&](size_t bytes) -> void* {
        off = (off + 15) & ~(size_t)15;
        void* p = ws + off;
        off += bytes;
        return p;
    };
    int*   firstcp    = (int*)  carve((size_t)BB * NSEG * sizeof(int));
    float* segsum     = (float*)carve((size_t)BB * NSEG * sizeof(float));
    float* segcnt     = (float*)carve((size_t)BB * NSEG * sizeof(float));
    float* sums       = (float*)carve((size_t)BB * 4 * sizeof(float));
    int*   posCnt     = (int*)  carve((size_t)BB * sizeof(int));
    int*   compactCnt = (int*)  carve((size_t)BB * sizeof(int));
    float* repsum     = (float*)carve((size_t)BB * sizeof(float));
    float* norms      = (float*)carve((size_t)BB * NI * sizeof(float));
    float* cpEmb      = (float*)carve((size_t)BB * NI * DD * sizeof(float));

    hipLaunchKernelGGL(oc_init, dim3(128), dim3(256), 0, stream,
                       firstcp, posCnt, compactCnt, sums, segsum, segcnt, repsum, norms, cpEmb);

    hipLaunchKernelGGL(oc_pass1, dim3((NN + 255) / 256, BB), dim3(256), 0, stream,
                       beta, embed, sid, iscp, firstcp, posCnt, compactCnt, sums, cpEmb, norms);

    hipLaunchKernelGGL(oc_attr, dim3(128, BB), dim3(256), 0, stream,
                       embed, sid, firstcp, segsum, segcnt);

    // 256 tiles per batch, 8 waves (tiles) per 256-thread block -> 32 blocks x
    hipLaunchKernelGGL(oc_rep, dim3(32, BB), dim3(256), 0, stream,
                       cpEmb, norms, posCnt, repsum);

    hipLaunchKernelGGL(oc_final, dim3(1), dim3(256), 0, stream,
                       sums, posCnt, firstcp, segsum, segcnt, repsum, out);
}